// IndRNN_onlyrecurrent_15487652069798
// MI455X (gfx1250) — compile-verified
//
#include <hip/hip_runtime.h>

typedef unsigned int v4u __attribute__((ext_vector_type(4)));
typedef int          v4i __attribute__((ext_vector_type(4)));
typedef int          v8i __attribute__((ext_vector_type(8)));

// Problem constants (match reference)
constexpr int LEN    = 2048;
constexpr int BATCH  = 64;
constexpr int HIDDEN = 512;
constexpr int COLS   = BATCH * HIDDEN;   // 32768 floats per timestep row

// Tiling: deep TDM pipeline. In-flight fraction = (NB-1)/NB of staged LDS,
// so many small buffers > few big ones. 2 waves * 10 bufs * 16 ts * 512B = 160KB.
constexpr int T      = 16;               // timesteps per TDM tile
constexpr int NB     = 10;               // LDS buffers per wave (NB-1 tiles in flight)
constexpr int WCOLS  = 128;              // columns per wave (32 lanes x float4)
constexpr int BLOCK  = 64;               // 2 waves per block
constexpr int WAVES  = BLOCK / 32;
constexpr int BCOLS  = BLOCK * 4;        // 256 columns per block
constexpr int NTILES = LEN / T;          // 128

__global__ __launch_bounds__(BLOCK)
void indrnn_recurrent_kernel(const float* __restrict__ x,
                             const float* __restrict__ w,
                             float* __restrict__ out) {
    // [wave][buffer][timestep][column] : 2*10*16*128*4 = 160 KB static LDS
    __shared__ float sh[WAVES][NB][T][WCOLS];

    const int lane    = threadIdx.x & 31;
    const int wave    = threadIdx.x >> 5;
    const int colBase = blockIdx.x * BCOLS + wave * WCOLS; // wave's first column
    const int col     = colBase + lane * 4;                // this thread's 4 columns

    // Per-thread recurrent weights: weight index = flat column % HIDDEN
    const float4 wv = *(const float4*)(w + (col & (HIDDEN - 1)));

    // ---- Tensor DMA descriptor, group 1 (constant per wave) ----
    // bits[17:16] data_size=2 (4B); tensor_dim0=WCOLS; tensor_dim1=T;
    // tile_dim0=WCOLS; tile_dim1=T; tile_dim2=0 (2D); dim0 stride = COLS elems.
    v8i g1;
    g1[0] = (2 << 16);            // workgroup_mask=0, data_size=4B, no pad/iter
    g1[1] = (WCOLS << 16);        // atomic_barrier_addr=0 | tensor_dim0[15:0]
    g1[2] = (T << 16);            // tensor_dim0[31:16]=0 | tensor_dim1[15:0]
    g1[3] = (WCOLS << 16);        // tensor_dim1[31:16]=0 | tile_dim0
    g1[4] = T;                    // tile_dim1 | tile_dim2=0
    g1[5] = COLS;                 // tensor_dim0_stride[31:0]
    g1[6] = 0;                    // stride0[47:32]=0 | stride1[15:0]=0 (2D)
    g1[7] = 0;
    const v4i gz4 = {0, 0, 0, 0};                 // groups 2/3 unused (2D tile)
    const v8i gz8 = {0, 0, 0, 0, 0, 0, 0, 0};     // extension group (clang-23 form)

    auto issue_tile = [&](int tile, int buf) {
        const float* gp = x + (size_t)tile * T * COLS + colBase;
        unsigned long long ga = (unsigned long long)(const void*)gp;
        // Flat LDS pointer: byte offset within workgroup LDS is addr[31:0]
        unsigned int la = (unsigned int)(unsigned long long)(const void*)
                          &sh[wave][buf][0][0];
        v4u g0;
        g0[0] = 1u;                                            // count=1 (valid)
        g0[1] = la;                                            // lds_addr
        g0[2] = (unsigned int)ga;                              // global_addr[31:0]
        g0[3] = (unsigned int)((ga >> 32) & 0x1FFFFFFu)        // global_addr[56:32]
              | (2u << 30);                                    // type=2 ("image")
        __builtin_amdgcn_tensor_load_to_lds(g0, g1, gz4, gz4, gz8, 0);
    };

    // Prime the pipeline: NB-1 tiles in flight per wave
    #pragma unroll
    for (int i = 0; i < NB - 1; ++i) issue_tile(i, i);

    float4 h = make_float4(0.f, 0.f, 0.f, 0.f);
    float* outp = out + col;

    int bufC = 0;           // buffer consumed at iteration i   (= i % NB)
    int bufF = NB - 1;      // buffer filled  at iteration i    (= (i+NB-1) % NB)
    for (int i = 0; i < NTILES; ++i) {
        // Oldest tile (i) complete when TENSORcnt <= NB-2 (in-order per wave)
        __builtin_amdgcn_s_wait_tensorcnt(NB - 2);
        asm volatile("" ::: "memory");  // keep LDS reads below the wait

        // Refill the buffer consumed at iteration i-1
        if (i + NB - 1 < NTILES) issue_tile(i + NB - 1, bufF);

        const float4* xs = (const float4*)&sh[wave][bufC][0][lane * 4];
        #pragma unroll
        for (int t = 0; t < T; ++t) {
            const float4 xv = xs[t * (WCOLS / 4)];
            h.x = fmaxf(fmaf(h.x, wv.x, xv.x), 0.f);
            h.y = fmaxf(fmaf(h.y, wv.y, xv.y), 0.f);
            h.z = fmaxf(fmaf(h.z, wv.z, xv.z), 0.f);
            h.w = fmaxf(fmaf(h.w, wv.w, xv.w), 0.f);
            *(float4*)(outp + (size_t)(i * T + t) * COLS) = h;
        }
        asm volatile("" ::: "memory");

        if (++bufC == NB) bufC = 0;
        if (++bufF == NB) bufF = 0;
    }
}

extern "C" void kernel_launch(void* const* d_in, const int* in_sizes, int n_in,
                              void* d_out, int out_size, void* d_ws, size_t ws_size,
                              hipStream_t stream) {
    const float* x = (const float*)d_in[0];           // [LEN, BATCH, HIDDEN] f32
    const float* w = (const float*)d_in[1];           // [HIDDEN] f32
    float* out = (float*)d_out;                       // [LEN, BATCH, HIDDEN] f32
    (void)in_sizes; (void)n_in; (void)out_size; (void)d_ws; (void)ws_size;

    const int grid = COLS / BCOLS;                    // 128 blocks x 64 threads
    indrnn_recurrent_kernel<<<grid, BLOCK, 0, stream>>>(x, w, out);
}